// AttentionLayer_27676769256201
// MI455X (gfx1250) — compile-verified
//
#include <hip/hip_runtime.h>
#include <math.h>

#define B 32
#define S 4096
#define H 768

typedef __attribute__((ext_vector_type(2))) float v2f;
typedef __attribute__((ext_vector_type(4))) float v4f;
typedef __attribute__((ext_vector_type(8))) float v8f;

// ---------------------------------------------------------------------------
// Pass 1: scores[b,s] = (hidden[b,s,:] . attn_w + attn_b) * (1 + 2*boost[b,s])
//         masked with -inf where attention_mask == 0.
// One wave32 per token, float4 coalesced loads. 8 tokens per 256-thread block.
// ---------------------------------------------------------------------------
__global__ __launch_bounds__(256) void scores_kernel(
    const float* __restrict__ hs, const int* __restrict__ mask,
    const int* __restrict__ boost, const float* __restrict__ attn_w,
    const float* __restrict__ attn_b, float* __restrict__ scores) {
  int wave = threadIdx.x >> 5;
  int lane = threadIdx.x & 31;
  int t = blockIdx.x * 8 + wave;  // token index in [0, B*S)
  const float* row = hs + (size_t)t * H;
  float sum = 0.f;
#pragma unroll
  for (int c = 0; c < 6; ++c) {   // 6 * 128 = 768
    int h = c * 128 + lane * 4;
    v4f x = *(const v4f*)(row + h);
    v4f w = *(const v4f*)(attn_w + h);
    sum += x.x * w.x + x.y * w.y + x.z * w.z + x.w * w.w;
  }
#pragma unroll
  for (int off = 16; off > 0; off >>= 1) sum += __shfl_down(sum, off, 32);
  if (lane == 0) {
    float sc = (sum + attn_b[0]) * (1.0f + 2.0f * (float)boost[t]);
    if (mask[t] == 0) sc = -INFINITY;
    scores[t] = sc;
  }
}

// ---------------------------------------------------------------------------
// Pass 2: softmax over the sequence dim. One block per batch row.
// ---------------------------------------------------------------------------
__global__ __launch_bounds__(256) void softmax_kernel(
    const float* __restrict__ scores, float* __restrict__ weights) {
  __shared__ float red[256];
  int b = blockIdx.x;
  const float* sc = scores + (size_t)b * S;
  float m = -INFINITY;
  for (int s = threadIdx.x; s < S; s += 256) m = fmaxf(m, sc[s]);
  red[threadIdx.x] = m;
  __syncthreads();
  for (int o = 128; o > 0; o >>= 1) {
    if (threadIdx.x < o) red[threadIdx.x] = fmaxf(red[threadIdx.x], red[threadIdx.x + o]);
    __syncthreads();
  }
  m = red[0];
  __syncthreads();
  float z = 0.f;
  for (int s = threadIdx.x; s < S; s += 256) z += expf(sc[s] - m);
  red[threadIdx.x] = z;
  __syncthreads();
  for (int o = 128; o > 0; o >>= 1) {
    if (threadIdx.x < o) red[threadIdx.x] += red[threadIdx.x + o];
    __syncthreads();
  }
  float inv = 1.0f / red[0];
  for (int s = threadIdx.x; s < S; s += 256)
    weights[(size_t)b * S + s] = expf(sc[s] - m) * inv;
}

// ---------------------------------------------------------------------------
// Zero-init the context accumulator (graph-safe, deterministic every call).
// ---------------------------------------------------------------------------
__global__ __launch_bounds__(256) void zero_kernel(float* __restrict__ p, int n) {
  int i = blockIdx.x * 256 + threadIdx.x;
  if (i < n) p[i] = 0.f;
}

// ---------------------------------------------------------------------------
// Pass 3: context[b,h] = sum_s weights[b,s] * hidden[b,s,h].
// Each block: one (b, 256-wide h-chunk, 512-long s-slice). h is the fast
// axis across threads -> fully coalesced streaming of hidden_states.
// Partial sums combined with atomicAdd. 32 * 3 * 8 = 768 blocks.
// ---------------------------------------------------------------------------
#define POOL_HC 256
#define POOL_NS 8
#define POOL_SL (S / POOL_NS)  // 512
__global__ __launch_bounds__(256) void pool_kernel(
    const float* __restrict__ hs, const float* __restrict__ weights,
    float* __restrict__ context) {
  const int nH = H / POOL_HC;  // 3
  int idx = blockIdx.x;
  int b = idx / (nH * POOL_NS);
  int r = idx % (nH * POOL_NS);
  int hc = r / POOL_NS;
  int sc = r % POOL_NS;
  int h = hc * POOL_HC + threadIdx.x;
  const float* base = hs + (size_t)b * S * H + (size_t)sc * POOL_SL * H + h;
  const float* w = weights + (size_t)b * S + sc * POOL_SL;
  float acc = 0.f;
  for (int s = 0; s < POOL_SL; ++s) acc = fmaf(w[s], base[(size_t)s * H], acc);
  atomicAdd(&context[b * H + h], acc);
}

// ---------------------------------------------------------------------------
// Pass 4: BatchNorm1d over the batch dim (training-mode batch stats, biased
// variance). One thread per feature; 32 values live in registers.
// ---------------------------------------------------------------------------
__global__ __launch_bounds__(768) void bn_kernel(
    const float* __restrict__ context, const float* __restrict__ gamma,
    const float* __restrict__ beta, float* __restrict__ cbn) {
  int h = threadIdx.x;
  float vals[B];
  float mean = 0.f;
#pragma unroll
  for (int b = 0; b < B; ++b) {
    vals[b] = context[b * H + h];
    mean += vals[b];
  }
  mean *= (1.0f / B);
  float var = 0.f;
#pragma unroll
  for (int b = 0; b < B; ++b) {
    float d = vals[b] - mean;
    var += d * d;
  }
  var *= (1.0f / B);
  float sc = rsqrtf(var + 1e-5f) * gamma[h];
  float bs = beta[h];
#pragma unroll
  for (int b = 0; b < B; ++b) cbn[b * H + h] = (vals[b] - mean) * sc + bs;
}

// ---------------------------------------------------------------------------
// Pass 5: out = relu(cbn @ fc_w.T + fc_b + cbn) via V_WMMA_F32_16X16X4_F32.
// M=32, N=768, K=768. One wave per 16x16 output tile, 96 tiles.
// A[m][k] = cbn[(m0+m)*H + k]          (16x4 fp32 fragment, 2 VGPRs)
// B[k][n] = fc_w[(n0+n)*H + k]  (fc_w.T) (4x16 fp32 fragment, 2 VGPRs)
// Fragment layout per ISA 7.12.2: lanes 0-15 carry K={k0,k0+1} in the two
// VGPRs, lanes 16-31 carry K={k0+2,k0+3}; each lane loads a contiguous float2.
// ---------------------------------------------------------------------------
__global__ __launch_bounds__(32) void fc_wmma_kernel(
    const float* __restrict__ cbn, const float* __restrict__ fcw,
    const float* __restrict__ fcb, float* __restrict__ out) {
  const int nT = H / 16;              // 48 tiles along N
  int tile = blockIdx.x;              // 0..95
  int m0 = (tile / nT) * 16;
  int n0 = (tile % nT) * 16;
  int lane = threadIdx.x;             // wave32, EXEC all ones
  int half = lane >> 4;               // 0: K pair {k0,k0+1}; 1: {k0+2,k0+3}
  int l = lane & 15;

  const float* arow = cbn + (size_t)(m0 + l) * H + half * 2;
  const float* brow = fcw + (size_t)(n0 + l) * H + half * 2;

  v8f c = {};
  for (int k0 = 0; k0 < H; k0 += 4) {
    v2f a = *(const v2f*)(arow + k0);
    v2f bb = *(const v2f*)(brow + k0);
    // 8 args: (neg_a, A, neg_b, B, c_mod, C, reuse_a, reuse_b)
    c = __builtin_amdgcn_wmma_f32_16x16x4_f32(false, a, false, bb, (short)0, c,
                                              false, false);
  }

  int col = n0 + l;
  float fb = fcb[col];
#pragma unroll
  for (int i = 0; i < 8; ++i) {
    int row = m0 + half * 8 + i;      // C/D layout: VGPR i = row i (+8 for hi half)
    float v = c[i] + fb + cbn[row * H + col];
    out[row * H + col] = fmaxf(v, 0.0f);
  }
}

extern "C" void kernel_launch(void* const* d_in, const int* in_sizes, int n_in,
                              void* d_out, int out_size, void* d_ws, size_t ws_size,
                              hipStream_t stream) {
  (void)in_sizes; (void)n_in; (void)out_size; (void)ws_size;
  const float* hs     = (const float*)d_in[0];  // [B,S,H]
  const int*   mask   = (const int*)d_in[1];    // [B,S]
  const int*   boost  = (const int*)d_in[2];    // [B,S]
  const float* attn_w = (const float*)d_in[3];  // [H]
  const float* attn_b = (const float*)d_in[4];  // scalar
  const float* fc_w   = (const float*)d_in[5];  // [H,H]
  const float* fc_b   = (const float*)d_in[6];  // [H]
  const float* gamma  = (const float*)d_in[7];  // [H]
  const float* beta   = (const float*)d_in[8];  // [H]
  float* out = (float*)d_out;                   // [B,H]

  float* scores  = (float*)d_ws;        // B*S
  float* weights = scores + B * S;      // B*S
  float* context = weights + B * S;     // B*H
  float* cbn     = context + B * H;     // B*H

  scores_kernel<<<(B * S) / 8, 256, 0, stream>>>(hs, mask, boost, attn_w,
                                                 attn_b, scores);
  softmax_kernel<<<B, 256, 0, stream>>>(scores, weights);
  zero_kernel<<<(B * H + 255) / 256, 256, 0, stream>>>(context, B * H);
  pool_kernel<<<B * (H / POOL_HC) * POOL_NS, 256, 0, stream>>>(hs, weights,
                                                               context);
  bn_kernel<<<1, H, 0, stream>>>(context, gamma, beta, cbn);
  fc_wmma_kernel<<<(B / 16) * (H / 16), 32, 0, stream>>>(cbn, fc_w, fc_b, out);
}